// BottlenectedAttention_31610959299297
// MI455X (gfx1250) — compile-verified
//
#include <hip/hip_runtime.h>
#include <hip/hip_bf16.h>

// Problem constants (from reference)
#define EDIM 512
#define HIDD 640
#define NHEAD 10
#define DKH 64
#define LTOT 2054          // 1024 + 1 + 4 + 1024 + 1
#define NBATCH 4
#define NCLS 527
#define VROW 1029          // La + 1 + BTNK

typedef __attribute__((ext_vector_type(2))) float v2f;
typedef __attribute__((ext_vector_type(8))) float v8f;

// ---------------------------------------------------------------------------
// Kernel 1: build feats = concat(...) + positional encoding (fp32)
// ---------------------------------------------------------------------------
__global__ void build_feats_kernel(const float* __restrict__ af,
                                   const float* __restrict__ vf,
                                   const float* __restrict__ atok,
                                   const float* __restrict__ vtok,
                                   const float* __restrict__ btok,
                                   float* __restrict__ feats, int total) {
  int idx = blockIdx.x * blockDim.x + threadIdx.x;
  if (idx >= total) return;
  int e = idx % EDIM;
  int t = idx / EDIM;
  int l = t % LTOT;
  int b = t / LTOT;
  float src;
  if (l < 1024)        src = af[((size_t)b * 1024 + l) * EDIM + e];
  else if (l == 1024)  src = atok[e];
  else if (l < 1029)   src = btok[(l - 1025) * EDIM + e];
  else if (l < 2053)   src = vf[((size_t)b * 1024 + (l - 1029)) * EDIM + e];
  else                 src = vtok[e];
  // pe: div = exp(-(2i)*ln(10000)/E); even->sin, odd->cos
  int i2 = e & ~1;
  float freq = __expf((float)i2 * (-9.210340371976184f / (float)EDIM));
  float ang = (float)l * freq;
  float pe = (e & 1) ? cosf(ang) : sinf(ang);
  feats[idx] = src + pe;
}

// ---------------------------------------------------------------------------
// Kernel 2: q = feats@Wq + bq ; v = feats@Wv + bv  via V_WMMA_F32_16X16X4_F32
// One wave computes a 64(M) x 16(N) tile of BOTH q and v:
//   - 4 M-subtiles share each B fragment (amortizes W loads 4x)
//   - q and v share each A fragment
// => 8 WMMAs per 8 vmem issues in the K-loop.
// ---------------------------------------------------------------------------
__global__ __launch_bounds__(32)
void qv_gemm_kernel(const float* __restrict__ feats,
                    const float* __restrict__ Wq, const float* __restrict__ biasq,
                    const float* __restrict__ Wv, const float* __restrict__ biasv,
                    float* __restrict__ qout, float* __restrict__ vout) {
  const int lane  = threadIdx.x;
  const int l16   = lane & 15;
  const int half  = lane >> 4;       // 0: K 0..1, 1: K 2..3
  const int ntile = blockIdx.x;      // 0..39  (N = 640/16)
  const int mtile = blockIdx.y;      // 0..32  (64-row supertiles over 2054)
  const int b     = blockIdx.z;      // 0..3
  const int n     = ntile * 16 + l16;

  const float* __restrict__ arow[4];
#pragma unroll
  for (int s = 0; s < 4; ++s) {
    int mA = mtile * 64 + s * 16 + l16;
    if (mA >= LTOT) mA = 0;          // clamp: garbage rows never stored
    arow[s] = feats + ((size_t)b * LTOT + mA) * EDIM;
  }

  const float* __restrict__ wq = Wq + n;
  const float* __restrict__ wv = Wv + n;

  v8f accq[4], accv[4];
#pragma unroll
  for (int s = 0; s < 4; ++s) {
    accq[s] = (v8f){0.f, 0.f, 0.f, 0.f, 0.f, 0.f, 0.f, 0.f};
    accv[s] = (v8f){0.f, 0.f, 0.f, 0.f, 0.f, 0.f, 0.f, 0.f};
  }

#pragma unroll 2
  for (int k0 = 0; k0 < EDIM; k0 += 4) {
    const int ka = k0 + 2 * half;
    // B fragments (4x16 f32): rows ka, ka+1 / ka+2, ka+3 split across halves
    v2f fq, fv;
    fq.x = wq[(size_t)ka * HIDD];
    fq.y = wq[(size_t)(ka + 1) * HIDD];
    fv.x = wv[(size_t)ka * HIDD];
    fv.y = wv[(size_t)(ka + 1) * HIDD];
    // prefetch W rows ~32 K ahead (clamped in-range; global_prefetch_b8)
    const int kpf = (ka + 32 < EDIM) ? (ka + 32) : (EDIM - 2);
    __builtin_prefetch(wq + (size_t)kpf * HIDD, 0, 0);
    __builtin_prefetch(wv + (size_t)kpf * HIDD, 0, 0);
    // A fragments (16x4 f32) for the 4 M-subtiles
    v2f a0 = *(const v2f*)(arow[0] + ka);
    v2f a1 = *(const v2f*)(arow[1] + ka);
    v2f a2 = *(const v2f*)(arow[2] + ka);
    v2f a3 = *(const v2f*)(arow[3] + ka);
    accq[0] = __builtin_amdgcn_wmma_f32_16x16x4_f32(false, a0, false, fq, (short)0, accq[0], false, false);
    accq[1] = __builtin_amdgcn_wmma_f32_16x16x4_f32(false, a1, false, fq, (short)0, accq[1], false, false);
    accq[2] = __builtin_amdgcn_wmma_f32_16x16x4_f32(false, a2, false, fq, (short)0, accq[2], false, false);
    accq[3] = __builtin_amdgcn_wmma_f32_16x16x4_f32(false, a3, false, fq, (short)0, accq[3], false, false);
    accv[0] = __builtin_amdgcn_wmma_f32_16x16x4_f32(false, a0, false, fv, (short)0, accv[0], false, false);
    accv[1] = __builtin_amdgcn_wmma_f32_16x16x4_f32(false, a1, false, fv, (short)0, accv[1], false, false);
    accv[2] = __builtin_amdgcn_wmma_f32_16x16x4_f32(false, a2, false, fv, (short)0, accv[2], false, false);
    accv[3] = __builtin_amdgcn_wmma_f32_16x16x4_f32(false, a3, false, fv, (short)0, accv[3], false, false);
  }

  const float bqn = biasq[n];
  const float bvn = biasv[n];
#pragma unroll
  for (int s = 0; s < 4; ++s) {
#pragma unroll
    for (int r = 0; r < 8; ++r) {
      const int m = mtile * 64 + s * 16 + r + 8 * half;  // C/D layout
      if (m < LTOT) {
        const size_t o = ((size_t)b * LTOT + m) * HIDD + n;
        qout[o] = accq[s][r] + bqn;
        vout[o] = accv[s][r] + bvn;
      }
    }
  }
}

// ---------------------------------------------------------------------------
// Kernel 3: k at the two CLS rows only: kcls[(b*2+r)*640 + h]
// ---------------------------------------------------------------------------
__global__ void kcls_kernel(const float* __restrict__ feats,
                            const float* __restrict__ Wk,
                            const float* __restrict__ bk,
                            float* __restrict__ kcls) {
  const int mrow = blockIdx.x;                 // 0 -> row 0, 1 -> row 1029
  const int b = blockIdx.y;
  const int m = mrow ? VROW : 0;
  const float* fr = feats + ((size_t)b * LTOT + m) * EDIM;
  for (int h = threadIdx.x; h < HIDD; h += blockDim.x) {
    float acc = bk[h];
    for (int e = 0; e < EDIM; ++e) acc += fr[e] * Wk[(size_t)e * HIDD + h];
    kcls[(b * 2 + mrow) * HIDD + h] = acc;
  }
}

// ---------------------------------------------------------------------------
// Kernel 4: attention for one CLS row + layernorm. 640 threads/block,
// heads processed in chunks of 5 to keep static LDS < 64 KB.
// ---------------------------------------------------------------------------
#define HCHUNK 5
__global__ __launch_bounds__(640)
void attn_kernel(const float* __restrict__ kcls,
                 const float* __restrict__ q, const float* __restrict__ v,
                 const float* __restrict__ ln_g, const float* __restrict__ ln_b,
                 float* __restrict__ cls) {
  __shared__ float s_logit[HCHUNK * LTOT];   // 41080 B
  __shared__ float s_k[HIDD];
  __shared__ float s_max[HCHUNK];
  __shared__ float s_sum[HCHUNK];
  __shared__ float s_red[1024];
  __shared__ float s_red2[1024];

  const int tid  = threadIdx.x;
  const int mrow = blockIdx.x;
  const int b    = blockIdx.y;
  const int row  = b * 2 + mrow;
  if (tid < HIDD) s_k[tid] = kcls[row * HIDD + tid];
  __syncthreads();

  const float* __restrict__ qb = q + (size_t)b * LTOT * HIDD;
  const float* __restrict__ vb = v + (size_t)b * LTOT * HIDD;
  const int myh = tid >> 6;       // 0..9
  const int myd = tid & 63;
  float outval = 0.f;

  for (int hc = 0; hc < NHEAD; hc += HCHUNK) {
    // logits[hl, n] = (k_cls[h] . q[n,h]) / 8
    for (int idx = tid; idx < HCHUNK * LTOT; idx += 640) {
      const int hl = idx / LTOT;
      const int n  = idx - hl * LTOT;
      const int h  = hc + hl;
      const float* qp = qb + (size_t)n * HIDD + h * DKH;
      const float* kp = s_k + h * DKH;
      float acc = 0.f;
      for (int d = 0; d < DKH; ++d) acc += kp[d] * qp[d];
      s_logit[hl * LTOT + n] = acc * 0.125f;
    }
    __syncthreads();
    // per-head max & sum: one wave32 per head
    const int wave = tid >> 5, lw = tid & 31;
    if (wave < HCHUNK) {
      const float* lg = s_logit + wave * LTOT;
      float mx = -1e30f;
      for (int n = lw; n < LTOT; n += 32) mx = fmaxf(mx, lg[n]);
      for (int off = 16; off > 0; off >>= 1) mx = fmaxf(mx, __shfl_xor(mx, off, 32));
      float sm = 0.f;
      for (int n = lw; n < LTOT; n += 32) sm += __expf(lg[n] - mx);
      for (int off = 16; off > 0; off >>= 1) sm += __shfl_xor(sm, off, 32);
      if (lw == 0) { s_max[wave] = mx; s_sum[wave] = sm; }
    }
    __syncthreads();
    // probs in place
    for (int idx = tid; idx < HCHUNK * LTOT; idx += 640) {
      const int hl = idx / LTOT;
      s_logit[idx] = __expf(s_logit[idx] - s_max[hl]) / s_sum[hl];
    }
    __syncthreads();
    // out[h*64+d] = sum_n p[n,h] * v[n, h*64+d]
    if (myh >= hc && myh < hc + HCHUNK) {
      const float* lg = s_logit + (myh - hc) * LTOT;
      const float* vp = vb + myh * DKH + myd;
      float acc = 0.f;
      for (int n = 0; n < LTOT; ++n) acc += lg[n] * vp[(size_t)n * HIDD];
      outval = acc;
    }
    __syncthreads();
  }

  // layernorm over 640 values (one per thread)
  s_red[tid]  = outval;
  s_red2[tid] = outval * outval;
  if (tid < 1024 - HIDD) { s_red[HIDD + tid] = 0.f; s_red2[HIDD + tid] = 0.f; }
  __syncthreads();
  for (int s = 512; s > 0; s >>= 1) {
    if (tid < s) { s_red[tid] += s_red[tid + s]; s_red2[tid] += s_red2[tid + s]; }
    __syncthreads();
  }
  const float mean = s_red[0] * (1.0f / HIDD);
  const float var  = s_red2[0] * (1.0f / HIDD) - mean * mean;
  const float inv  = rsqrtf(var + 1e-5f);
  cls[row * HIDD + tid] = (outval - mean) * inv * ln_g[tid] + ln_b[tid];
}

// ---------------------------------------------------------------------------
// Kernel 5: out[b,c] = 0.5*(cls_a@Wap + bap + cls_v@Wvp + bvp)
// ---------------------------------------------------------------------------
__global__ void pred_kernel(const float* __restrict__ cls,
                            const float* __restrict__ Wap, const float* __restrict__ bap,
                            const float* __restrict__ Wvp, const float* __restrict__ bvp,
                            float* __restrict__ out) {
  __shared__ float s_a[HIDD];
  __shared__ float s_v[HIDD];
  const int b = blockIdx.x;
  for (int i = threadIdx.x; i < HIDD; i += blockDim.x) {
    s_a[i] = cls[(b * 2 + 0) * HIDD + i];
    s_v[i] = cls[(b * 2 + 1) * HIDD + i];
  }
  __syncthreads();
  for (int c = threadIdx.x; c < NCLS; c += blockDim.x) {
    float acc = bap[c] + bvp[c];
    for (int h = 0; h < HIDD; ++h)
      acc += s_a[h] * Wap[(size_t)h * NCLS + c] + s_v[h] * Wvp[(size_t)h * NCLS + c];
    out[b * NCLS + c] = 0.5f * acc;
  }
}

// ---------------------------------------------------------------------------
extern "C" void kernel_launch(void* const* d_in, const int* in_sizes, int n_in,
                              void* d_out, int out_size, void* d_ws, size_t ws_size,
                              hipStream_t stream) {
  (void)in_sizes; (void)n_in; (void)out_size; (void)ws_size;
  const float* audio_feat = (const float*)d_in[0];
  const float* video_feat = (const float*)d_in[1];
  const float* audio_tok  = (const float*)d_in[2];
  const float* video_tok  = (const float*)d_in[3];
  const float* btnk_toks  = (const float*)d_in[4];
  const float* Wk   = (const float*)d_in[5];
  const float* bk   = (const float*)d_in[6];
  const float* Wq   = (const float*)d_in[7];
  const float* bq_  = (const float*)d_in[8];
  const float* Wv   = (const float*)d_in[9];
  const float* bv_  = (const float*)d_in[10];
  const float* ln_g = (const float*)d_in[11];
  const float* ln_b = (const float*)d_in[12];
  const float* Wap  = (const float*)d_in[13];
  const float* bap  = (const float*)d_in[14];
  const float* Wvp  = (const float*)d_in[15];
  const float* bvp  = (const float*)d_in[16];
  float* out = (float*)d_out;

  // workspace layout (fp32)
  float* ws    = (float*)d_ws;
  float* feats = ws;                                         // 4*2054*512
  float* qbuf  = feats + (size_t)NBATCH * LTOT * EDIM;       // 4*2054*640
  float* vbuf  = qbuf + (size_t)NBATCH * LTOT * HIDD;        // 4*2054*640
  float* kcls  = vbuf + (size_t)NBATCH * LTOT * HIDD;        // 8*640
  float* cls   = kcls + 2 * NBATCH * HIDD;                   // 8*640

  const int total = NBATCH * LTOT * EDIM;
  build_feats_kernel<<<(total + 255) / 256, 256, 0, stream>>>(
      audio_feat, video_feat, audio_tok, video_tok, btnk_toks, feats, total);

  qv_gemm_kernel<<<dim3(HIDD / 16, (LTOT + 63) / 64, NBATCH), 32, 0, stream>>>(
      feats, Wq, bq_, Wv, bv_, qbuf, vbuf);

  kcls_kernel<<<dim3(2, NBATCH), 256, 0, stream>>>(feats, Wk, bk, kcls);

  attn_kernel<<<dim3(2, NBATCH), 640, 0, stream>>>(kcls, qbuf, vbuf, ln_g, ln_b, cls);

  pred_kernel<<<NBATCH, 256, 0, stream>>>(cls, Wap, bap, Wvp, bvp, out);
}